// OBS_loss_18047452577914
// MI455X (gfx1250) — compile-verified
//
#include <hip/hip_runtime.h>

#define DDIM 9
#define NB1  4096
#define BS   256

typedef __attribute__((ext_vector_type(2))) float v2f;
typedef __attribute__((ext_vector_type(8))) float v8f;

// Pass 1: stream Y/pred/nan_array (non-temporal: 540MB >> 192MB L2),
// accumulate per-row  sum(((Y-P)*invnorm)^2) / count(!isnan)  and tree-reduce
// per block into d_ws. Row-per-thread, grid-stride; deterministic.
__global__ __launch_bounds__(BS) void obs_loss_pass1(
    const float* __restrict__ Y, const float* __restrict__ P,
    const float* __restrict__ A, const float* __restrict__ norm,
    float* __restrict__ ws, long long nRows)
{
    float invn[DDIM];
#pragma unroll
    for (int d = 0; d < DDIM; ++d) invn[d] = 1.0f / norm[d];

    float acc = 0.0f;
    const long long stride = (long long)gridDim.x * BS;
    for (long long r = (long long)blockIdx.x * BS + threadIdx.x; r < nRows; r += stride) {
        const float* y = Y + r * DDIM;
        const float* p = P + r * DDIM;
        const float* a = A + r * DDIM;
        float ss  = 0.0f;
        float cnt = 0.0f;
#pragma unroll
        for (int d = 0; d < DDIM; ++d) {
            float t = (__builtin_nontemporal_load(y + d) -
                       __builtin_nontemporal_load(p + d)) * invn[d];
            ss = fmaf(t, t, ss);
            float nv = __builtin_nontemporal_load(a + d);
            cnt += (nv == nv) ? 1.0f : 0.0f;   // !isnan
        }
        acc += ss / cnt;   // cnt==0 -> +inf, matching the reference exactly
    }

    __shared__ float sdata[BS];
    sdata[threadIdx.x] = acc;
    __syncthreads();
#pragma unroll
    for (int s = BS / 2; s > 0; s >>= 1) {
        if (threadIdx.x < s) sdata[threadIdx.x] += sdata[threadIdx.x + s];
        __syncthreads();
    }
    if (threadIdx.x == 0) ws[blockIdx.x] = sdata[0];
}

// Pass 2: one block reduces the per-block partials; final 32-lane sum via
// V_WMMA_F32_16X16X4_F32 (A VGPR0 = lane values, VGPR1 = 0, B = ones, C = 0):
//   D[m][n] = v_m + v_{m+16};  lane0 d[0..7] = D[0..7][0], lane16 d[0..7] = D[8..15][0]
// so (sum of d[] in lane0) + (sum of d[] in lane16) = sum of all 32 lane values.
__global__ __launch_bounds__(BS) void obs_loss_pass2(
    const float* __restrict__ ws, int nParts, float* __restrict__ out, float invN)
{
    float acc = 0.0f;
    for (int i = threadIdx.x; i < nParts; i += BS) acc += ws[i];

    __shared__ float sdata[BS];
    sdata[threadIdx.x] = acc;
    __syncthreads();

    if (threadIdx.x < 32) {          // wave 0 fully active -> EXEC all-1s for WMMA
        float v = 0.0f;
#pragma unroll
        for (int j = 0; j < BS / 32; ++j) v += sdata[threadIdx.x + 32 * j];

#if __has_builtin(__builtin_amdgcn_wmma_f32_16x16x4_f32)
        v2f a; a[0] = v;    a[1] = 0.0f;   // A[m][0]=v_m (lanes 0-15), A[m][2]=v_{m+16}
        v2f b; b[0] = 1.0f; b[1] = 1.0f;   // B = all-ones (layout-agnostic)
        v8f c = {};
        v8f dmat = __builtin_amdgcn_wmma_f32_16x16x4_f32(
            /*neg_a=*/false, a, /*neg_b=*/false, b,
            /*c_mod=*/(short)0, c, /*reuse_a=*/false, /*reuse_b=*/false);
        float s = dmat[0] + dmat[1] + dmat[2] + dmat[3] +
                  dmat[4] + dmat[5] + dmat[6] + dmat[7];
        float s_hi = __shfl(s, 16, 32);    // lane16 half of the total
        if (threadIdx.x == 0) out[0] = (s + s_hi) * invN;
#else
        // Shuffle fallback (wave32)
        for (int off = 16; off > 0; off >>= 1) v += __shfl_down(v, off, 32);
        if (threadIdx.x == 0) out[0] = v * invN;
#endif
    }
}

extern "C" void kernel_launch(void* const* d_in, const int* in_sizes, int n_in,
                              void* d_out, int out_size, void* d_ws, size_t ws_size,
                              hipStream_t stream) {
    const float* Y    = (const float*)d_in[0];
    const float* P    = (const float*)d_in[1];
    const float* A    = (const float*)d_in[2];
    const float* norm = (const float*)d_in[3];
    const long long nRows = (long long)in_sizes[0] / DDIM;

    float* ws = (float*)d_ws;
    int nb = NB1;
    if ((size_t)nb * sizeof(float) > ws_size)          // ws_size fixed per run ->
        nb = (int)(ws_size / sizeof(float));           // still deterministic

    obs_loss_pass1<<<nb, BS, 0, stream>>>(Y, P, A, norm, ws, nRows);
    obs_loss_pass2<<<1, BS, 0, stream>>>(ws, nb, (float*)d_out, 1.0f / (float)nRows);
}